// CharCRFDependency_33904471834770
// MI455X (gfx1250) — compile-verified
//
#include <hip/hip_runtime.h>
#include <hip/hip_bf16.h>
#include <stdint.h>

#define L    128
#define LP   129          // padded row stride (bank-conflict-free column reads)
#define BATCH 256
#define MINV (-1e30f)

typedef float v2f __attribute__((ext_vector_type(2)));
typedef float v8f __attribute__((ext_vector_type(8)));

// LDS byte offset of a generic pointer that points into __shared__ memory.
__device__ __forceinline__ unsigned lds_off_u32(const void* p) {
  return (unsigned)(unsigned long long)(__attribute__((address_space(3))) const void*)p;
}

// ---------------------------------------------------------------------------
// Kernel 1: Eisner inside DP. One workgroup per batch element.
// LDS: s_c[128][129], s_i[128][129], arc[128][128]  (~193 KB of 320 KB WGP LDS)
// Spans are distributed over 8 waves x G lane-groups of S lanes,
// S = next_pow2(min(w,32)), so small widths still use all 256 lanes.
// ---------------------------------------------------------------------------
__global__ __launch_bounds__(256)
void eisner_inside_kernel(const float* __restrict__ scores,
                          const unsigned char* __restrict__ mask,
                          float* __restrict__ logZ_out)
{
  extern __shared__ float lds[];
  float* s_c = lds;             // 128*129 = 16512 floats
  float* s_i = lds + 16512;     // 128*129
  float* arc = lds + 33024;     // 128*128 = scores[b] row-major [dep][head]
  __shared__ int lens_sh;

  const int tid  = threadIdx.x;
  const int lane = tid & 31;
  const int wv   = tid >> 5;    // 8 waves
  const int b    = blockIdx.x;

  // ---- async DMA: scores[b] (64 KB) -> LDS arc[], tracked by ASYNCcnt ----
  {
    const char* gbase   = (const char*)(scores + (size_t)b * L * L);
    unsigned    ldsbase = lds_off_u32(arc);
#pragma unroll
    for (int k = 0; k < 16; ++k) {
      unsigned off = (unsigned)((k * 256 + tid) * 16);
      unsigned dst = ldsbase + off;
      asm volatile("global_load_async_to_lds_b128 %0, %1, %2"
                   :: "v"(dst), "v"(off), "s"(gbase)
                   : "memory");
    }
  }

  // ---- init charts to MIN (s_c diagonal = 0) while DMA is in flight ----
  for (int idx = tid; idx < 33024; idx += 256) {
    float v = (idx < 16512 && (idx % 130) == 0) ? 0.0f : MINV;  // d*129+d = 130*d
    lds[idx] = v;
  }
  // lens[b] = sum(mask[b,:]) computed by wave 0
  if (wv == 0) {
    const unsigned char* mrow = mask + (size_t)b * L;
    int c = 0;
    for (int l = lane; l < L; l += 32) c += (int)mrow[l];
#pragma unroll
    for (int off = 16; off > 0; off >>= 1) c += __shfl_xor(c, off, 32);
    if (lane == 0) lens_sh = c;
  }

  asm volatile("s_wait_asynccnt 0x0" ::: "memory");
  __syncthreads();

  // ---- width loop ----
  for (int w = 1; w < L; ++w) {
    const int n = L - w;

    // group geometry for this width (wave-uniform)
    int lg = 0;
    while ((1 << lg) < w && lg < 5) ++lg;
    const int S      = 1 << lg;       // lanes per span
    const int G      = 32 >> lg;      // spans per wave
    const int sub    = lane & (S - 1);
    const int grp    = lane >> lg;
    const int stride = 8 * G;         // spans per block step
    const int ibase  = wv * G + grp;

    // Phase 1: incomplete spans. il = LSE_j( s_c[i][i+j] + s_c[i+w][i+1+j] )
    for (int i = ibase; i < n; i += stride) {
      const float* rowA = s_c + i * LP + i;
      const float* rowB = s_c + (i + w) * LP + (i + 1);
      float m = MINV, s = 0.0f;
      for (int j = sub; j < w; j += S) {
        float v = rowA[j] + rowB[j];
        if (v > m) { s = s * __expf(m - v) + 1.0f; m = v; }
        else       { s += __expf(v - m); }
      }
      for (int off = S >> 1; off > 0; off >>= 1) {   // stays inside the group
        float m2 = __shfl_xor(m, off, 32);
        float s2 = __shfl_xor(s, off, 32);
        float mm = fmaxf(m, m2);
        s = s * __expf(m - mm) + s2 * __expf(m2 - mm);
        m = mm;
      }
      if (sub == 0) {
        float il = m + __logf(s);
        // s_arc[x,y] = scores[b][y][x]  ->  arc[y*L + x]
        s_i[(i + w) * LP + i]  = il + arc[i * L + (i + w)];
        s_i[i * LP + (i + w)]  = il + arc[(i + w) * L + i];
      }
    }
    __syncthreads();

    // Phase 2: complete spans.
    // cl = LSE_j( s_c[i+j][i]   + s_i[i+w][i+j]   )
    // cr = LSE_j( s_i[i][i+1+j] + s_c[i+1+j][i+w] )
    for (int i = ibase; i < n; i += stride) {
      float ml = MINV, sl = 0.0f, mr = MINV, sr = 0.0f;
      for (int j = sub; j < w; j += S) {
        float vl = s_c[(i + j) * LP + i] + s_i[(i + w) * LP + (i + j)];
        float vr = s_i[i * LP + (i + 1 + j)] + s_c[(i + 1 + j) * LP + (i + w)];
        if (vl > ml) { sl = sl * __expf(ml - vl) + 1.0f; ml = vl; }
        else         { sl += __expf(vl - ml); }
        if (vr > mr) { sr = sr * __expf(mr - vr) + 1.0f; mr = vr; }
        else         { sr += __expf(vr - mr); }
      }
      for (int off = S >> 1; off > 0; off >>= 1) {
        float m2 = __shfl_xor(ml, off, 32), s2 = __shfl_xor(sl, off, 32);
        float mm = fmaxf(ml, m2);
        sl = sl * __expf(ml - mm) + s2 * __expf(m2 - mm); ml = mm;
        m2 = __shfl_xor(mr, off, 32); s2 = __shfl_xor(sr, off, 32);
        mm = fmaxf(mr, m2);
        sr = sr * __expf(mr - mm) + s2 * __expf(m2 - mm); mr = mm;
      }
      if (sub == 0) {
        s_c[(i + w) * LP + i] = ml + __logf(sl);
        float cr = mr + __logf(sr);
        if (i == 0 && lens_sh != w) cr = MINV;   // root-span length mask
        s_c[i * LP + (i + w)] = cr;
      }
    }
    __syncthreads();
  }

  if (tid == 0) logZ_out[b] = s_c[lens_sh];      // s_c[0][lens]
}

// ---------------------------------------------------------------------------
// Kernel 2: gold-score + mask-count + logZ reduction -> loss.
// Gold sum accumulated on the matrix unit: D = A x ones(4x16) + C via
// V_WMMA_F32_16X16X4_F32. Gather loads are unconditional (target always
// in-range) and masked by multiply, so the loads pipeline with no exec
// branches.
// ---------------------------------------------------------------------------
__global__ __launch_bounds__(256)
void crf_finalize_kernel(const float* __restrict__ scores,
                         const unsigned char* __restrict__ mask,
                         const int* __restrict__ target,
                         const float* __restrict__ logZ_ws,
                         float* __restrict__ out)
{
  __shared__ float pg[8];
  __shared__ int   pc[8];
  __shared__ float pz[8];

  const int tid  = threadIdx.x;
  const int lane = tid & 31;
  const int wv   = tid >> 5;
  const int base = wv * (BATCH * L / 8);        // 4096 pairs per wave

  v8f acc = {};
  v2f onesv; onesv.x = 1.0f; onesv.y = 1.0f;
  int cnt = 0;

  for (int k = 0; k < 64; ++k) {                // 64 iters * 64 elems = 4096
    int p0 = base + k * 64 + 2 * lane;
    int p1 = p0 + 1;
    int m0 = (int)mask[p0];
    int m1 = (int)mask[p1];
    float v0 = scores[((size_t)p0 << 7) + target[p0]] * (float)m0;
    float v1 = scores[((size_t)p1 << 7) + target[p1]] * (float)m1;
    cnt += m0 + m1;
    v2f a; a.x = v0; a.y = v1;
    // D[m][n] = sum_k A[m][k] + C[m][n]  (B = ones): matrix-unit reduction
    acc = __builtin_amdgcn_wmma_f32_16x16x4_f32(false, a, false, onesv,
                                                (short)0, acc, false, false);
  }

  // sum over all lanes/VGPRs of D = 16 * (true sum)  (16 identical columns)
  float lane_sum = acc[0] + acc[1] + acc[2] + acc[3] +
                   acc[4] + acc[5] + acc[6] + acc[7];
#pragma unroll
  for (int off = 16; off > 0; off >>= 1) {
    lane_sum += __shfl_xor(lane_sum, off, 32);
    cnt      += __shfl_xor(cnt, off, 32);
  }

  // per-thread logZ partial (BATCH == blockDim.x == 256)
  float z = logZ_ws[tid];
#pragma unroll
  for (int off = 16; off > 0; off >>= 1) z += __shfl_xor(z, off, 32);

  if (lane == 0) {
    pg[wv] = lane_sum * (1.0f / 16.0f);
    pc[wv] = cnt;
    pz[wv] = z;
  }
  __syncthreads();

  if (tid == 0) {
    float G = 0.0f, Z = 0.0f; int C = 0;
#pragma unroll
    for (int i = 0; i < 8; ++i) { G += pg[i]; Z += pz[i]; C += pc[i]; }
    out[0] = (Z - G) / (float)C;
  }
}

// ---------------------------------------------------------------------------
extern "C" void kernel_launch(void* const* d_in, const int* in_sizes, int n_in,
                              void* d_out, int out_size, void* d_ws, size_t ws_size,
                              hipStream_t stream) {
  (void)in_sizes; (void)n_in; (void)out_size; (void)ws_size;
  const float*         scores = (const float*)d_in[0];
  const unsigned char* mask   = (const unsigned char*)d_in[1];  // jax bool = 1 byte
  const int*           target = (const int*)d_in[2];
  float* out     = (float*)d_out;
  float* logZ_ws = (float*)d_ws;     // BATCH floats

  const size_t lds_bytes = (size_t)(2 * 128 * 129 + 128 * 128) * sizeof(float); // 197632
  eisner_inside_kernel<<<BATCH, 256, lds_bytes, stream>>>(scores, mask, logZ_ws);
  crf_finalize_kernel<<<1, 256, 0, stream>>>(scores, mask, target, logZ_ws, out);
}